// ASAP_Pool_55860344652298
// MI455X (gfx1250) — compile-verified
//
#include <hip/hip_runtime.h>
#include <math.h>

// ---------------- problem constants ----------------
#define NN   8192          // nodes
#define CC   256           // channels
#define EE   262144        // input edges
#define ET   (EE + NN)     // edges incl. self loops = 270336
#define KSEL 4096          // k = ceil(0.5*N)
#define NEG_SLOPE 0.2f

typedef __attribute__((ext_vector_type(2))) float v2f;
typedef __attribute__((ext_vector_type(8))) float v8f;

#if __has_builtin(__builtin_amdgcn_tensor_load_to_lds) && __has_builtin(__builtin_amdgcn_s_wait_tensorcnt)
#define HAVE_TDM 1
typedef unsigned tdm_u32x4 __attribute__((ext_vector_type(4)));
typedef int      tdm_i32x8 __attribute__((ext_vector_type(8)));
typedef int      tdm_i32x4 __attribute__((ext_vector_type(4)));
#else
#define HAVE_TDM 0
#endif

// edge accessors: ei = edge_index flat [2,E]; e >= EE are the appended self loops
__device__ __forceinline__ int eRow(const int* ei, int e) { return e < EE ? ei[e]      : e - EE; }
__device__ __forceinline__ int eCol(const int* ei, int e) { return e < EE ? ei[EE + e] : e - EE; }

// monotone-uint mapping so float max can use u32 atomic max
__device__ __forceinline__ unsigned f2mono(float f) {
  unsigned u = __float_as_uint(f);
  return (u & 0x80000000u) ? ~u : (u | 0x80000000u);
}
__device__ __forceinline__ float mono2f(unsigned u) {
  return (u & 0x80000000u) ? __uint_as_float(u & 0x7FFFFFFFu) : __uint_as_float(~u);
}

// ---------------- fp32 WMMA GEMM: D = A(MxK) @ B(KxN) (+bias) ----------------
// block = 256 threads (8 waves) computing a 128(M) x 16(N) strip.
// The K x 16 panel of B is DMA'd into LDS once per block with the Tensor Data
// Mover (tensor_load_to_lds + s_wait_tensorcnt), then each wave runs the
// 16x16x4 fp32 WMMA K-loop with A from global (float2/lane) and B from LDS.
__global__ __launch_bounds__(256) void k_gemm_wmma(
    const float* __restrict__ A, const float* __restrict__ B,
    const float* __restrict__ bias, float* __restrict__ D,
    int M, int Nn, int K)
{
  __shared__ float ldsB[CC * 16];        // K x 16 panel (K <= CC), 16 KB

  int ntn = Nn >> 4;
  int bm  = blockIdx.x / ntn;            // 128-row M strip
  int nt  = blockIdx.x - bm * ntn;
  if (bm >= (M >> 7)) return;            // uniform
  int n0 = nt << 4;

#if HAVE_TDM
  if (threadIdx.x < 32) {                // wave 0 issues the TDM descriptor
    unsigned long long ga = (unsigned long long)(uintptr_t)(B + n0);
    unsigned lds_off = (unsigned)(uintptr_t)&ldsB[0];   // addr[31:0] == LDS byte offset
    // group0: count=1 | lds_addr | global_addr[56:0] | type=2
    tdm_u32x4 g0;
    g0[0] = 1u;
    g0[1] = lds_off;
    g0[2] = (unsigned)(ga & 0xFFFFFFFFull);
    g0[3] = (unsigned)((ga >> 32) & 0x1FFFFFFull) | 0x80000000u;
    // group1: data_size=4B; tensor_dim0=Nn; tensor_dim1=K; tile_dim0=16;
    //         tile_dim1=K; tensor_dim0_stride=Nn
    tdm_i32x8 g1;
    g1[0] = 2 << 16;                                        // data_size = 4 bytes
    g1[1] = (int)(((unsigned)Nn & 0xFFFFu) << 16);          // tensor_dim0 lo16
    g1[2] = (int)(((unsigned)Nn >> 16) |
                  (((unsigned)K & 0xFFFFu) << 16));         // dim0 hi16 | tensor_dim1 lo16
    g1[3] = (int)(((unsigned)K >> 16) | (16u << 16));       // dim1 hi16 | tile_dim0 = 16
    g1[4] = K & 0xFFFF;                                     // tile_dim1 = K rows
    g1[5] = Nn;                                             // tensor_dim0_stride lo32
    g1[6] = 0;
    g1[7] = 0;
    tdm_i32x4 gz4 = {0, 0, 0, 0};
    tdm_i32x8 gz8 = {0, 0, 0, 0, 0, 0, 0, 0};
    __builtin_amdgcn_tensor_load_to_lds(g0, g1, gz4, gz4, gz8, 0);
    __builtin_amdgcn_s_wait_tensorcnt(0);
  }
  __syncthreads();
#else
  for (int i = threadIdx.x; i < K * 16; i += 256) {
    int k = i >> 4, n = i & 15;
    ldsB[i] = B[(size_t)k * Nn + n0 + n];
  }
  __syncthreads();
#endif

  int wv   = threadIdx.x >> 5;
  int lane = threadIdx.x & 31;
  int m0   = ((bm << 3) + wv) << 4;
  int lh = lane >> 4, lm = lane & 15;

  v8f c = {0.f, 0.f, 0.f, 0.f, 0.f, 0.f, 0.f, 0.f};
  const float* arow = A + (size_t)(m0 + lm) * K + 2 * lh;   // A[m][k0 + 2lh ..]
  const float* brow = ldsB + (2 * lh) * 16 + lm;            // B[k0 + 2lh][n]
  for (int k0 = 0; k0 < K; k0 += 4) {
    v2f a, b;
    a[0] = arow[k0];
    a[1] = arow[k0 + 1];
    b[0] = brow[k0 * 16];
    b[1] = brow[(k0 + 1) * 16];
    c = __builtin_amdgcn_wmma_f32_16x16x4_f32(false, a, false, b, (short)0, c, false, false);
  }
  float bv = bias ? bias[n0 + lm] : 0.f;
#pragma unroll
  for (int v = 0; v < 8; ++v)
    D[(size_t)(m0 + v + 8 * lh) * Nn + n0 + lm] = c[v] + bv;
}

// ---------------- degree (by row, float) + in-degree histogram (by col, int) ----------------
__global__ void k_deg_cnt(const int* ei, float* deg, int* cnt) {
  int e = blockIdx.x * blockDim.x + threadIdx.x;
  if (e >= ET) return;
  atomicAdd(&deg[eRow(ei, e)], 1.0f);
  atomicAdd(&cnt[eCol(ei, e)], 1);
}

__global__ void k_dinv(const float* deg, float* dinv) {
  int n = blockIdx.x * blockDim.x + threadIdx.x;
  if (n >= NN) return;
  float d = deg[n];
  dinv[n] = d > 0.f ? rsqrtf(d) : 0.f;
}

// x_pool[row] += dinv[row]*dinv[col] * XW[col]   (block = edge, thread = channel)
__global__ __launch_bounds__(256) void k_scatter_pool(const int* ei, const float* dinv,
                                                      const float* XW, float* x_pool) {
  int e = blockIdx.x, c = threadIdx.x;
  int r = eRow(ei, e), cl = eCol(ei, e);
  float nrm = dinv[r] * dinv[cl];
  atomicAdd(&x_pool[(size_t)r * CC + c], nrm * XW[(size_t)cl * CC + c]);
}

__global__ void k_add_bias(float* x_pool, const float* bgcn) {
  int i = blockIdx.x * blockDim.x + threadIdx.x;
  if (i >= NN * CC) return;
  x_pool[i] += bgcn[i & (CC - 1)];
}

// X_q[row] = segmax x_pool[col]  (mono-uint atomic max)
__global__ __launch_bounds__(256) void k_xq_max(const int* ei, const float* x_pool, unsigned* xq_u) {
  int e = blockIdx.x, c = threadIdx.x;
  int r = eRow(ei, e), cl = eCol(ei, e);
  atomicMax(&xq_u[(size_t)r * CC + c], f2mono(x_pool[(size_t)cl * CC + c]));
}

__global__ void k_mono_arr(unsigned* buf, int n) {
  int i = blockIdx.x * blockDim.x + threadIdx.x;
  if (i >= n) return;
  float f = mono2f(buf[i]);
  ((float*)buf)[i] = f;
}

// a1[n] = Mq[n]·Wa[:C], a2[n] = x_pool[n]·Wa[C:]
__global__ __launch_bounds__(256) void k_node_att(const float* Mq, const float* x_pool,
                                                  const float* Wa, float* a1, float* a2) {
  __shared__ float sh[256];
  int n = blockIdx.x, t = threadIdx.x;
  float v1 = Mq[(size_t)n * CC + t] * Wa[t];
  float v2 = x_pool[(size_t)n * CC + t] * Wa[CC + t];
  sh[t] = v1; __syncthreads();
  for (int s = 128; s > 0; s >>= 1) { if (t < s) sh[t] += sh[t + s]; __syncthreads(); }
  if (t == 0) a1[n] = sh[0];
  __syncthreads();
  sh[t] = v2; __syncthreads();
  for (int s = 128; s > 0; s >>= 1) { if (t < s) sh[t] += sh[t + s]; __syncthreads(); }
  if (t == 0) a2[n] = sh[0];
}

// raw = leaky_relu(a1[row]+a2[col]+ba); row-wise max via mono atomic
__global__ void k_raw(const int* ei, const float* a1, const float* a2, const float* ba,
                      float* raw, unsigned* m_u) {
  int e = blockIdx.x * blockDim.x + threadIdx.x;
  if (e >= ET) return;
  int r = eRow(ei, e), c = eCol(ei, e);
  float v = a1[r] + a2[c] + ba[0];
  v = v < 0.f ? v * NEG_SLOPE : v;
  raw[e] = v;
  atomicMax(&m_u[r], f2mono(v));
}

__global__ void k_exp(const int* ei, const float* raw, const unsigned* m_u,
                      float* expv, float* esum) {
  int e = blockIdx.x * blockDim.x + threadIdx.x;
  if (e >= ET) return;
  int r = eRow(ei, e);
  float ex = expf(raw[e] - mono2f(m_u[r]));
  expv[e] = ex;
  atomicAdd(&esum[r], ex);
}

__global__ void k_att(const int* ei, const float* expv, const float* esum, float* att) {
  int e = blockIdx.x * blockDim.x + threadIdx.x;
  if (e >= ET) return;
  att[e] = expv[e] / esum[eRow(ei, e)];
}

// out[row] += att * x[col]
__global__ __launch_bounds__(256) void k_out_scatter(const int* ei, const float* att,
                                                     const float* x, float* out) {
  int e = blockIdx.x, c = threadIdx.x;
  int r = eRow(ei, e), cl = eCol(ei, e);
  atomicAdd(&out[(size_t)r * CC + c], att[e] * x[(size_t)cl * CC + c]);
}

// per-node LEConv dots: h = out·Wle, l1 = out·W1 + b1, l2 = out·W2 + b2
__global__ __launch_bounds__(256) void k_le(const float* out, const float* Wle, const float* W1,
                                            const float* b1, const float* W2, const float* b2,
                                            float* h, float* l1, float* l2) {
  __shared__ float sh[256];
  int n = blockIdx.x, t = threadIdx.x;
  float o = out[(size_t)n * CC + t];
  sh[t] = o * Wle[t]; __syncthreads();
  for (int s = 128; s > 0; s >>= 1) { if (t < s) sh[t] += sh[t + s]; __syncthreads(); }
  if (t == 0) h[n] = sh[0];
  __syncthreads();
  sh[t] = o * W1[t]; __syncthreads();
  for (int s = 128; s > 0; s >>= 1) { if (t < s) sh[t] += sh[t + s]; __syncthreads(); }
  if (t == 0) l1[n] = sh[0] + b1[0];
  __syncthreads();
  sh[t] = o * W2[t]; __syncthreads();
  for (int s = 128; s > 0; s >>= 1) { if (t < s) sh[t] += sh[t + s]; __syncthreads(); }
  if (t == 0) l2[n] = sh[0] + b2[0];
}

__global__ void k_deg2_aggr(const int* ei, const float* h, float* deg2, float* aggr) {
  int e = blockIdx.x * blockDim.x + threadIdx.x;
  if (e >= ET) return;
  int r = eRow(ei, e), c = eCol(ei, e);
  if (r != c) {
    atomicAdd(&deg2[r], 1.0f);
    atomicAdd(&aggr[r], h[c]);
  }
}

__global__ void k_fitness(const float* deg2, const float* l1, const float* aggr,
                          const float* l2, float* fit) {
  int n = blockIdx.x * blockDim.x + threadIdx.x;
  if (n >= NN) return;
  float le = deg2[n] * l1[n] + aggr[n] + l2[n];
  fit[n] = 1.f / (1.f + expf(-le));
}

// exact lax.top_k order (descending, stable ties) via rank counting
__global__ __launch_bounds__(256) void k_topk(const float* f, int* perm, float* topv,
                                              int* n_idx, int* in_perm) {
  __shared__ int sh[256];
  int i = blockIdx.x, t = threadIdx.x;
  float fi = f[i];
  int cnt = 0;
  for (int j = t; j < NN; j += 256) {
    float fj = f[j];
    cnt += (fj > fi) || (fj == fi && j < i);
  }
  sh[t] = cnt; __syncthreads();
  for (int s = 128; s > 0; s >>= 1) { if (t < s) sh[t] += sh[t + s]; __syncthreads(); }
  if (t == 0) {
    int rank = sh[0];
    if (rank < KSEL) { perm[rank] = i; topv[rank] = fi; n_idx[i] = rank; in_perm[i] = 1; }
  }
}

__global__ __launch_bounds__(256) void k_xnew(const float* out, const int* perm,
                                              const float* topv, float* xnew) {
  int j = blockIdx.x, c = threadIdx.x;
  xnew[(size_t)j * CC + c] = out[(size_t)perm[j] * CC + c] * topv[j];
}

// single-thread exclusive scan over 8192 in-degrees (one-time, negligible)
__global__ void k_scan(const int* cnt, int* offs) {
  if (blockIdx.x == 0 && threadIdx.x == 0) {
    int acc = 0;
    for (int i = 0; i < NN; ++i) { offs[i] = acc; acc += cnt[i]; }
    offs[NN] = acc;
  }
}

__global__ void k_csr_fill(const int* ei, const int* offs, int* cursor, int* inEdge) {
  int e = blockIdx.x * blockDim.x + threadIdx.x;
  if (e >= ET) return;
  int c = eCol(ei, e);
  int pos = offs[c] + atomicAdd(&cursor[c], 1);
  inEdge[pos] = e;
}

// Ek[c1,c2] += att[e1]*att[e2] over edge pairs (e1 in in(row_e), e2 in in(col_e))
__global__ __launch_bounds__(64) void k_ek(const int* ei, const int* offs, const int* inEdge,
                                           const int* n_idx, const int* in_perm,
                                           const float* att, float* Ek) {
  int e = blockIdx.x;
  int r = eRow(ei, e), c = eCol(ei, e);
  int br = offs[r], dr = offs[r + 1] - br;
  int bc = offs[c], dc = offs[c + 1] - bc;
  __builtin_prefetch(&inEdge[br], 0, 3);        // global_prefetch_b8: in-edge lists
  __builtin_prefetch(&inEdge[bc], 0, 3);
  int total = dr * dc;
  for (int t = threadIdx.x; t < total; t += blockDim.x) {
    int i = t / dc, j = t - i * dc;
    int e1 = inEdge[br + i];
    int r1 = eRow(ei, e1);
    if (!in_perm[r1]) continue;
    int e2 = inEdge[bc + j];
    int r2 = eRow(ei, e2);
    if (!in_perm[r2]) continue;
    atomicAdd(&Ek[(size_t)n_idx[r1] * KSEL + n_idx[r2]], att[e1] * att[e2]);
  }
}

__global__ void k_ek_diag(float* Ek) {
  int j = blockIdx.x * blockDim.x + threadIdx.x;
  if (j >= KSEL) return;
  Ek[(size_t)j * KSEL + j] = 1.0f;
}

// ---------------- host orchestration ----------------
extern "C" void kernel_launch(void* const* d_in, const int* in_sizes, int n_in,
                              void* d_out, int out_size, void* d_ws, size_t ws_size,
                              hipStream_t stream)
{
  const float* x    = (const float*)d_in[0];
  const int*   ei   = (const int*)  d_in[1];
  const float* Wgcn = (const float*)d_in[2];
  const float* bgcn = (const float*)d_in[3];
  const float* Wq   = (const float*)d_in[4];
  const float* bq   = (const float*)d_in[5];
  const float* Wa   = (const float*)d_in[6];
  const float* ba   = (const float*)d_in[7];
  const float* Wle  = (const float*)d_in[8];
  const float* W1   = (const float*)d_in[9];
  const float* b1   = (const float*)d_in[10];
  const float* W2   = (const float*)d_in[11];
  const float* b2   = (const float*)d_in[12];

  // output layout: x_new [k*C] | Ek [k*k] | att [Et]
  float* xnewOut = (float*)d_out;
  float* ekOut   = xnewOut + (size_t)KSEL * CC;
  float* attOut  = ekOut + (size_t)KSEL * KSEL;

  // workspace carve-up
  char* ws = (char*)d_ws;
  size_t off = 0;
  auto alloc = [&](size_t bytes) -> void* {
    void* p = ws + off;
    off = (off + bytes + 255) & ~(size_t)255;
    return p;
  };
  float*    XW     = (float*)   alloc(sizeof(float) * (size_t)NN * CC);
  float*    x_pool = (float*)   alloc(sizeof(float) * (size_t)NN * CC);
  unsigned* xq     = (unsigned*)alloc(sizeof(unsigned) * (size_t)NN * CC); // mono -> float in place
  float*    Mq     = (float*)   alloc(sizeof(float) * (size_t)NN * CC);
  float*    outF   = (float*)   alloc(sizeof(float) * (size_t)NN * CC);
  float*    deg    = (float*)   alloc(sizeof(float) * NN);
  float*    dinv   = (float*)   alloc(sizeof(float) * NN);
  float*    a1     = (float*)   alloc(sizeof(float) * NN);
  float*    a2     = (float*)   alloc(sizeof(float) * NN);
  float*    raw    = (float*)   alloc(sizeof(float) * ET);
  float*    expv   = (float*)   alloc(sizeof(float) * ET);
  unsigned* m_u    = (unsigned*)alloc(sizeof(unsigned) * NN);
  float*    esum   = (float*)   alloc(sizeof(float) * NN);
  float*    h      = (float*)   alloc(sizeof(float) * NN);
  float*    l1     = (float*)   alloc(sizeof(float) * NN);
  float*    l2     = (float*)   alloc(sizeof(float) * NN);
  float*    deg2   = (float*)   alloc(sizeof(float) * NN);
  float*    aggr   = (float*)   alloc(sizeof(float) * NN);
  float*    fit    = (float*)   alloc(sizeof(float) * NN);
  int*      perm   = (int*)     alloc(sizeof(int) * KSEL);
  float*    topv   = (float*)   alloc(sizeof(float) * KSEL);
  int*      n_idx  = (int*)     alloc(sizeof(int) * NN);
  int*      inperm = (int*)     alloc(sizeof(int) * NN);
  int*      cnt    = (int*)     alloc(sizeof(int) * NN);
  int*      offs   = (int*)     alloc(sizeof(int) * (NN + 1));
  int*      cursor = (int*)     alloc(sizeof(int) * NN);
  int*      inEdge = (int*)     alloc(sizeof(int) * ET);
  (void)ws_size; (void)in_sizes; (void)n_in; (void)out_size;

  // zero every accumulator (deterministic across replays; graph-capture safe)
  (void)hipMemsetAsync(deg,    0, sizeof(float) * NN, stream);
  (void)hipMemsetAsync(cnt,    0, sizeof(int) * NN, stream);
  (void)hipMemsetAsync(cursor, 0, sizeof(int) * NN, stream);
  (void)hipMemsetAsync(x_pool, 0, sizeof(float) * (size_t)NN * CC, stream);
  (void)hipMemsetAsync(xq,     0, sizeof(unsigned) * (size_t)NN * CC, stream);
  (void)hipMemsetAsync(outF,   0, sizeof(float) * (size_t)NN * CC, stream);
  (void)hipMemsetAsync(m_u,    0, sizeof(unsigned) * NN, stream);
  (void)hipMemsetAsync(esum,   0, sizeof(float) * NN, stream);
  (void)hipMemsetAsync(deg2,   0, sizeof(float) * NN, stream);
  (void)hipMemsetAsync(aggr,   0, sizeof(float) * NN, stream);
  (void)hipMemsetAsync(n_idx,  0, sizeof(int) * NN, stream);
  (void)hipMemsetAsync(inperm, 0, sizeof(int) * NN, stream);
  (void)hipMemsetAsync(ekOut,  0, sizeof(float) * (size_t)KSEL * KSEL, stream);

  const int T = 256;
  int gET  = (ET + T - 1) / T;
  int gN   = (NN + T - 1) / T;
  int gNC  = (NN * CC + T - 1) / T;
  int gemmBlocks = (NN / 128) * (CC / 16);   // block = 128x16 strip, 8 waves

  // 1. degrees + in-degree histogram, symmetric norm
  k_deg_cnt<<<gET, T, 0, stream>>>(ei, deg, cnt);
  k_dinv<<<gN, T, 0, stream>>>(deg, dinv);

  // 2. XW = x @ W_gcn (TDM-staged B panel + WMMA fp32)
  k_gemm_wmma<<<gemmBlocks, T, 0, stream>>>(x, Wgcn, nullptr, XW, NN, CC, CC);

  // 3. GCN aggregate + bias
  k_scatter_pool<<<ET, T, 0, stream>>>(ei, dinv, XW, x_pool);
  k_add_bias<<<gNC, T, 0, stream>>>(x_pool, bgcn);

  // 4. X_q = segment_max(x_pool[col], row)
  k_xq_max<<<ET, T, 0, stream>>>(ei, x_pool, xq);
  k_mono_arr<<<gNC, T, 0, stream>>>(xq, NN * CC);

  // 5. Mq = X_q @ Wq + bq (WMMA fp32), then per-node attention scalars
  k_gemm_wmma<<<gemmBlocks, T, 0, stream>>>((const float*)xq, Wq, bq, Mq, NN, CC, CC);
  k_node_att<<<NN, T, 0, stream>>>(Mq, x_pool, Wa, a1, a2);

  // 6. edge scores + segment softmax -> att (written straight into d_out)
  k_raw<<<gET, T, 0, stream>>>(ei, a1, a2, ba, raw, m_u);
  k_exp<<<gET, T, 0, stream>>>(ei, raw, m_u, expv, esum);
  k_att<<<gET, T, 0, stream>>>(ei, expv, esum, attOut);

  // 7. out = seg_sum(x[col] * att, row)
  k_out_scatter<<<ET, T, 0, stream>>>(ei, attOut, x, outF);

  // 8. LEConv fitness
  k_le<<<NN, T, 0, stream>>>(outF, Wle, W1, b1, W2, b2, h, l1, l2);
  k_deg2_aggr<<<gET, T, 0, stream>>>(ei, h, deg2, aggr);
  k_fitness<<<gN, T, 0, stream>>>(deg2, l1, aggr, l2, fit);

  // 9. top-k + x_new
  k_topk<<<NN, T, 0, stream>>>(fit, perm, topv, n_idx, inperm);
  k_xnew<<<KSEL, T, 0, stream>>>(outF, perm, topv, xnewOut);

  // 10. in-edge CSR (by col) then sparse Ek = S^T A S
  k_scan<<<1, 1, 0, stream>>>(cnt, offs);
  k_csr_fill<<<gET, T, 0, stream>>>(ei, offs, cursor, inEdge);
  k_ek<<<ET, 64, 0, stream>>>(ei, offs, inEdge, n_idx, inperm, attOut, ekOut);
  k_ek_diag<<<(KSEL + T - 1) / T, T, 0, stream>>>(ekOut);
}